// SymmetryBreakingGNN_20950850470259
// MI455X (gfx1250) — compile-verified
//
#include <hip/hip_runtime.h>

#define GCN_CH 64

typedef __attribute__((ext_vector_type(2))) float v2f;
typedef __attribute__((ext_vector_type(8))) float v8f;

// ---------------------------------------------------------------------------
// Y[Nx64] = (RELU ? relu(X) : X) @ W[64x64]
// One wave per 16-row strip. 16 K-steps of V_WMMA_F32_16X16X4_F32 x 4 col tiles.
// A 16x4 f32 layout (ISA 7.12.2): VGPR0 = {K0 lanes0-15, K2 lanes16-31},
//                                 VGPR1 = {K1 lanes0-15, K3 lanes16-31}
// B 4x16 mirrors A's K split; D 16x16: VGPR r = {M=r lanes0-15, M=r+8 lanes16-31}
// ---------------------------------------------------------------------------
template <bool RELU>
__global__ __launch_bounds__(256)
void gcn_gemm64_wmma(const float* __restrict__ X,
                     const float* __restrict__ W,
                     float* __restrict__ Y,
                     int row_tiles)
{
    const int wave = threadIdx.x >> 5;
    const int lane = threadIdx.x & 31;
    const int tile = blockIdx.x * 8 + wave;        // per-wave uniform -> EXEC stays all-1s
    if (tile >= row_tiles) return;

    const int half = lane >> 4;                    // 0: lanes 0-15, 1: lanes 16-31
    const int l    = lane & 15;

    const float* xrow = X + ((long long)tile * 16 + l) * GCN_CH;

    v8f acc0 = {}, acc1 = {}, acc2 = {}, acc3 = {};

#pragma unroll
    for (int kk = 0; kk < 16; ++kk) {
        const int k0 = kk * 4 + half * 2;

        // A fragment: contiguous K pair -> single b64 load
        float ax = xrow[k0];
        float ay = xrow[k0 + 1];
        if (RELU) {
            ax = fmaxf(ax, 0.0f);
            ay = fmaxf(ay, 0.0f);
        }
        v2f a; a.x = ax; a.y = ay;

        const float* w0 = W + (long long)k0 * GCN_CH;        // row K0 of this step
        const float* w1 = w0 + GCN_CH;                       // row K1

        v2f b0; b0.x = w0[l];      b0.y = w1[l];
        v2f b1; b1.x = w0[16 + l]; b1.y = w1[16 + l];
        v2f b2; b2.x = w0[32 + l]; b2.y = w1[32 + l];
        v2f b3; b3.x = w0[48 + l]; b3.y = w1[48 + l];

        acc0 = __builtin_amdgcn_wmma_f32_16x16x4_f32(false, a, false, b0, (short)0, acc0, false, false);
        acc1 = __builtin_amdgcn_wmma_f32_16x16x4_f32(false, a, false, b1, (short)0, acc1, false, false);
        acc2 = __builtin_amdgcn_wmma_f32_16x16x4_f32(false, a, false, b2, (short)0, acc2, false, false);
        acc3 = __builtin_amdgcn_wmma_f32_16x16x4_f32(false, a, false, b3, (short)0, acc3, false, false);
    }

    // D layout: VGPR r -> row tile*16 + half*8 + r, col = 16*ct + l
    const long long base = ((long long)tile * 16 + half * 8) * GCN_CH;
#pragma unroll
    for (int r = 0; r < 8; ++r) {
        float* yrow = Y + base + (long long)r * GCN_CH + l;
        yrow[0]  = acc0[r];
        yrow[16] = acc1[r];
        yrow[32] = acc2[r];
        yrow[48] = acc3[r];
    }
}

// ---------------------------------------------------------------------------
// agg[n, c] = bias[c]  (deterministically re-initializes poisoned buffers)
// ---------------------------------------------------------------------------
__global__ __launch_bounds__(256)
void gcn_init_bias(float* __restrict__ out, const float* __restrict__ bias, long long total)
{
    long long t = (long long)blockIdx.x * blockDim.x + threadIdx.x;
    if (t < total) out[t] = bias[t & (GCN_CH - 1)];
}

// ---------------------------------------------------------------------------
// For each edge e: agg[dst[e], :] += h[src[e], :]
// 32 lanes per edge, float2 (2 channels) per lane -> 256B contiguous per edge.
// Edge indices are wave-uniform: readfirstlane scalarizes them so the gather
// load and the global_atomic_add_f32 use SGPR-base addressing. Atomics resolve
// in L2 (agg buffer is 25.6MB, L2-resident at 192MB).
// ---------------------------------------------------------------------------
__global__ __launch_bounds__(256)
void gcn_scatter_add(const float* __restrict__ h,
                     const int* __restrict__ src,
                     const int* __restrict__ dst,
                     float* __restrict__ agg,
                     int n_edges)
{
    long long t = (long long)blockIdx.x * blockDim.x + threadIdx.x;
    int e    = (int)(t >> 5);           // wave-uniform (blockDim multiple of 32)
    int lane = (int)(t & 31);
    if (e >= n_edges) return;

    int s = __builtin_amdgcn_readfirstlane(src[e]);
    int d = __builtin_amdgcn_readfirstlane(dst[e]);

    const float2* hp = (const float2*)(h + (long long)s * GCN_CH);
    float2 v = hp[lane];

    float* ap = agg + (long long)d * GCN_CH + lane * 2;
    atomicAdd(ap,     v.x);
    atomicAdd(ap + 1, v.y);
}

extern "C" void kernel_launch(void* const* d_in, const int* in_sizes, int n_in,
                              void* d_out, int out_size, void* d_ws, size_t ws_size,
                              hipStream_t stream)
{
    const float* v0 = (const float*)d_in[0];   // [N, 64]
    const int*   ei = (const int*)d_in[1];     // [2, E]
    const float* W1 = (const float*)d_in[2];   // [64, 64]
    const float* b1 = (const float*)d_in[3];   // [64]
    const float* W2 = (const float*)d_in[4];   // [64, 64]
    const float* b2 = (const float*)d_in[5];   // [64]
    float* out = (float*)d_out;                // [N, 64]

    const int n_nodes = in_sizes[0] / GCN_CH;
    const int n_edges = in_sizes[1] / 2;
    const int* src = ei;
    const int* dst = ei + n_edges;

    const long long n_feat = (long long)n_nodes * GCN_CH;

    // Workspace: h (25.6MB) | agg1 (25.6MB). h is reused for layer-2 GEMM output.
    float* h    = (float*)d_ws;
    float* agg1 = h + n_feat;

    const int row_tiles   = n_nodes / 16;                       // 6250
    const int gemm_blocks = (row_tiles + 7) / 8;
    const int init_blocks = (int)((n_feat + 255) / 256);
    const int scat_blocks = (int)(((long long)n_edges * 32 + 255) / 256);

    dim3 blk(256);

    // ---- Layer 1: agg1 = A @ (v0 @ W1) + b1 ----
    gcn_gemm64_wmma<false><<<gemm_blocks, blk, 0, stream>>>(v0, W1, h, row_tiles);
    gcn_init_bias        <<<init_blocks, blk, 0, stream>>>(agg1, b1, n_feat);
    gcn_scatter_add      <<<scat_blocks, blk, 0, stream>>>(h, src, dst, agg1, n_edges);

    // ---- Layer 2: out = A @ (relu(agg1) @ W2) + b2  (ReLU fused into GEMM A-load) ----
    gcn_gemm64_wmma<true><<<gemm_blocks, blk, 0, stream>>>(agg1, W2, h, row_tiles);
    gcn_init_bias        <<<init_blocks, blk, 0, stream>>>(out, b2, n_feat);
    gcn_scatter_add      <<<scat_blocks, blk, 0, stream>>>(h, src, dst, out, n_edges);
}